// SigmaModel_43181601194713
// MI455X (gfx1250) — compile-verified
//
#include <hip/hip_runtime.h>

typedef __attribute__((ext_vector_type(2))) float v2f;
typedef __attribute__((ext_vector_type(4))) float v4f;
typedef __attribute__((ext_vector_type(8))) float v8f;

// Dimensions from the reference
#define SDIM   32
#define ADIM   8
#define INDIM  40     // SDIM + ADIM
#define HDIM   64
#define TRI    528    // 32*33/2
#define DDIM   32

#define HPITCH 68     // H LDS row pitch (floats): 4*m spread -> all 64 banks hit by float2 A-frag loads
#define YPITCH 532    // Y LDS row pitch (floats): +8 rows lands +32 banks, no half-wave aliasing

__global__ __launch_bounds__(32) void sigma_mlp_kernel(
    const float* __restrict__ s,  const float* __restrict__ a,
    const float* __restrict__ W1, const float* __restrict__ b1,
    const float* __restrict__ W2, const float* __restrict__ b2,
    float* __restrict__ out)
{
    __shared__ float Hlds[16 * HPITCH];   //  4,352 B
    __shared__ float Ylds[16 * YPITCH];   // 34,048 B

    const int tid  = threadIdx.x;         // wave32, one wave per block
    const int l15  = tid & 15;
    const int hi   = tid >> 4;            // 0: K={0,1}, 1: K={2,3} of each K=4 step
    const int koff = hi * 2;
    const int g0   = blockIdx.x * 16;     // first batch row of this tile
    const int g    = g0 + l15;            // this lane's batch row (A-matrix M index)

    // ---------------- A fragments of X = [s | a]  (16 x 40, f32) ----------------
    // A 16x4 layout: lanes 0-15 hold M=0..15 {K0,K1}; lanes 16-31 hold {K2,K3}.
    v2f Af[10];
#pragma unroll
    for (int kb = 0; kb < 8; ++kb)
        Af[kb] = *(const v2f*)(s + (size_t)g * SDIM + 4 * kb + koff);
#pragma unroll
    for (int kb = 8; kb < 10; ++kb)
        Af[kb] = *(const v2f*)(a + (size_t)g * ADIM + 4 * (kb - 8) + koff);

    // ---------------- Layer 1: H[16x64] = relu(X @ W1^T + b1) ----------------
#pragma unroll
    for (int nb = 0; nb < 4; ++nb) {
        const int n = nb * 16 + l15;                 // hidden column (B-matrix N index)
        const float* wrow = W1 + n * INDIM;
        v2f Bf[10];
#pragma unroll
        for (int kb = 0; kb < 10; ++kb)              // B 4x16: lane holds {K=koff,koff+1} of col n
            Bf[kb] = *(const v2f*)(wrow + 4 * kb + koff);

        v8f c0 = {}; v8f c1 = {};                    // two chains for tensor-pipe ILP
#pragma unroll
        for (int kb = 0; kb < 10; kb += 2) {
            c0 = __builtin_amdgcn_wmma_f32_16x16x4_f32(false, Af[kb],     false, Bf[kb],     (short)0, c0, false, false);
            c1 = __builtin_amdgcn_wmma_f32_16x16x4_f32(false, Af[kb + 1], false, Bf[kb + 1], (short)0, c1, false, false);
        }
        const float b1v = b1[n];
#pragma unroll
        for (int r = 0; r < 8; ++r) {                // D layout: VGPR r -> row r (+8 for hi half)
            float h = c0[r] + c1[r] + b1v;
            h = h > 0.0f ? h : 0.0f;
            Hlds[(r + hi * 8) * HPITCH + nb * 16 + l15] = h;
        }
    }
    __syncthreads();

    // A fragments of H (16 x 64), loaded once from LDS (bank-conflict-free)
    v2f Ha[16];
#pragma unroll
    for (int kb = 0; kb < 16; ++kb)
        Ha[kb] = *(const v2f*)&Hlds[l15 * HPITCH + 4 * kb + koff];

    // ---------------- Layer 2: Y[16x528] = H @ W2^T + b2 ----------------
    for (int nb = 0; nb < 33; ++nb) {
        const int t = nb * 16 + l15;                 // tri column index
        const float* wrow = W2 + t * HDIM;
        v2f Bf[16];
#pragma unroll
        for (int kb = 0; kb < 16; ++kb)
            Bf[kb] = *(const v2f*)(wrow + 4 * kb + koff);

        v8f c0 = {}; v8f c1 = {};
#pragma unroll
        for (int kb = 0; kb < 16; kb += 2) {
            c0 = __builtin_amdgcn_wmma_f32_16x16x4_f32(false, Ha[kb],     false, Bf[kb],     (short)0, c0, false, false);
            c1 = __builtin_amdgcn_wmma_f32_16x16x4_f32(false, Ha[kb + 1], false, Bf[kb + 1], (short)0, c1, false, false);
        }
        const float b2v = b2[t];
#pragma unroll
        for (int r = 0; r < 8; ++r)
            Ylds[(r + hi * 8) * YPITCH + t] = c0[r] + c1[r] + b2v;
    }
    __syncthreads();

    // ---------------- Expand tri -> symmetric sigma, coalesced 512B/instr stores ----------------
#pragma unroll 1
    for (int bl = 0; bl < 16; ++bl) {
        const float* Yrow = &Ylds[bl * YPITCH];
        float* orow = out + (size_t)(g0 + bl) * (DDIM * DDIM);
#pragma unroll
        for (int c = 0; c < 8; ++c) {
            const int flat = c * 128 + tid * 4;      // wave covers 512B contiguous per c
            const int i  = flat >> 5;
            const int j0 = flat & 31;
            v4f v;
#pragma unroll
            for (int e = 0; e < 4; ++e) {
                const int j  = j0 + e;
                const int ii = i < j ? i : j;
                const int jj = i < j ? j : i;
                const int tt = ii * (65 - ii) / 2 + (jj - ii);  // triu (row-major, incl diag)
                float f = Yrow[tt];
                if (i == j) f = __expf(f);           // exp on diagonal (sym fix cancels there)
                v[e] = f;
            }
            *(v4f*)(orow + flat) = v;
        }
    }
}

extern "C" void kernel_launch(void* const* d_in, const int* in_sizes, int n_in,
                              void* d_out, int out_size, void* d_ws, size_t ws_size,
                              hipStream_t stream) {
    const float* s  = (const float*)d_in[0];
    const float* a  = (const float*)d_in[1];
    const float* W1 = (const float*)d_in[2];
    const float* b1 = (const float*)d_in[3];
    const float* W2 = (const float*)d_in[4];
    const float* b2 = (const float*)d_in[5];
    float* out = (float*)d_out;

    const int batch = in_sizes[0] / SDIM;   // 131072
    const int tiles = batch / 16;           // 8192 waves, one per 32-thread block
    sigma_mlp_kernel<<<tiles, 32, 0, stream>>>(s, a, W1, b1, W2, b2, out);
}